// SparseTopologicalCrossAttn_32444182954025
// MI455X (gfx1250) — compile-verified
//
#include <hip/hip_runtime.h>
#include <hip/hip_bf16.h>
#include <math.h>

typedef float v2f __attribute__((ext_vector_type(2)));
typedef float v8f __attribute__((ext_vector_type(8)));

// ---------------------------------------------------------------------------
// WMMA f32 GEMM: D[M,N] = epilogue(A[M,K] @ B[K,N]) via V_WMMA_F32_16X16X4_F32.
// Compile-time K and strides -> all global loads use immediate offsets.
// Each wave computes RT stacked 16x16 C tiles (64x16 for RT=4), so the
// B fragment is reused across RT WMMAs per k-step.
//   MODE 0: D = X
//   MODE 1: D = sigmoid(X + bias[col]) * Mul[row,col]   (gating fusion)
// A-frag layout (ISA 7.12.2, 32-bit A 16x4): lanes 0-15 hold K={0,1},
// lanes 16-31 hold K={2,3}; B rows striped across lanes within a VGPR.
// ---------------------------------------------------------------------------
template<int K, int LDA, int LDB, int LDD, int LDM, int MODE, int RT>
__global__ __launch_bounds__(128) void gemm_wmma(
    const float* __restrict__ A, const float* __restrict__ B,
    const float* __restrict__ bias, float* __restrict__ D,
    const float* __restrict__ Mul, int M, int N)
{
    const int wave = threadIdx.x >> 5;
    const int lane = threadIdx.x & 31;
    const int half = lane >> 4;          // 0: lanes 0-15, 1: lanes 16-31
    const int lm   = lane & 15;

    const int row0 = blockIdx.x * (16 * RT);
    const int col0 = (blockIdx.y * 4 + wave) * 16;
    if (col0 >= N) return;               // wave-uniform: EXEC all-ones for WMMA

    const float* Ap[RT];
    #pragma unroll
    for (int j = 0; j < RT; ++j) {
        int r = min(row0 + 16 * j + lm, M - 1);
        Ap[j] = A + r * LDA + 2 * half;
    }
    const float* Bp = B + (2 * half) * LDB + col0 + lm;

    v8f acc[RT];
    #pragma unroll
    for (int j = 0; j < RT; ++j) acc[j] = (v8f){};

    #pragma unroll 4
    for (int k = 0; k < K; k += 4) {
        v2f b;
        b.x = Bp[k * LDB];                       // B[k+2h][col]
        b.y = Bp[(k + 1) * LDB];                 // B[k+1+2h][col]
        #pragma unroll
        for (int j = 0; j < RT; ++j) {
            v2f a = *(const v2f*)(Ap[j] + k);    // A[row][k+2h .. k+1+2h]
            acc[j] = __builtin_amdgcn_wmma_f32_16x16x4_f32(
                false, a, false, b, (short)0, acc[j], false, false);
        }
    }

    const int c = col0 + lm;
    #pragma unroll
    for (int j = 0; j < RT; ++j) {
        #pragma unroll
        for (int i = 0; i < 8; ++i) {
            int r = row0 + 16 * j + i + half * 8;   // C layout: VGPR i -> M=i / M=8+i
            if (r >= M) continue;
            float x = acc[j][i];
            if (MODE == 1) {
                float s = 1.0f / (1.0f + __expf(-(x + bias[c])));
                x = s * Mul[r * LDM + c];
            }
            D[r * LDD + c] = x;
        }
    }
}

// ---------------------------------------------------------------------------
// Small-N GEMM for the head-bias projections (N = 8 heads, K = 128).
// ---------------------------------------------------------------------------
__global__ void head_bias_gemm(const float* __restrict__ X,
                               const float* __restrict__ W,
                               float* __restrict__ out, int M)
{
    int idx = blockIdx.x * blockDim.x + threadIdx.x;
    if (idx >= M * 8) return;
    int r = idx >> 3, h = idx & 7;
    const float4* x = (const float4*)(X + (size_t)r * 128);
    float s = 0.f;
    #pragma unroll 8
    for (int i = 0; i < 32; ++i) {
        float4 v = x[i];
        s += v.x * W[(4 * i + 0) * 8 + h] + v.y * W[(4 * i + 1) * 8 + h] +
             v.z * W[(4 * i + 2) * 8 + h] + v.w * W[(4 * i + 3) * 8 + h];
    }
    out[idx] = s;
}

__global__ void fill_f32(float* p, float v, int n)
{
    int i = blockIdx.x * blockDim.x + threadIdx.x;
    if (i < n) p[i] = v;
}

// Monotonic float atomic-max: int max for >=0, uint min for <0 (init = -inf).
__device__ __forceinline__ void atomicMaxF(float* addr, float v)
{
    if (v >= 0.f) atomicMax((int*)addr, __float_as_int(v));
    else          atomicMin((unsigned int*)addr, __float_as_uint(v));
}

// Pass 1: per-(edge,head) scaled dot + biases; track segment max.
__global__ void edge_scores(const float* __restrict__ Q, const float* __restrict__ KV,
                            const float* __restrict__ bt, const float* __restrict__ bs,
                            const int* __restrict__ it, const int* __restrict__ is,
                            float* __restrict__ scores, float* mx, int nE, float scale)
{
    int idx = blockIdx.x * blockDim.x + threadIdx.x;
    if (idx >= nE * 8) return;
    int e = idx >> 3, h = idx & 7;
    int t = it[e], s = is[e];
    const float4* q = (const float4*)(Q + (size_t)t * 128 + h * 16);
    const float4* k = (const float4*)(KV + (size_t)s * 256 + h * 16);   // K half
    float d = 0.f;
    #pragma unroll
    for (int i = 0; i < 4; ++i) {
        float4 a = q[i], b = k[i];
        d += a.x * b.x + a.y * b.y + a.z * b.z + a.w * b.w;
    }
    float sc = d * scale + bt[t * 8 + h] + bs[s * 8 + h];
    scores[idx] = sc;
    atomicMaxF(mx + t * 8 + h, sc);
}

// Pass 2: exp(score - segmax), segment sum (scores overwritten with exp).
__global__ void edge_exp(float* __restrict__ scores, const float* __restrict__ mx,
                         const int* __restrict__ it, float* sm, int nE)
{
    int idx = blockIdx.x * blockDim.x + threadIdx.x;
    if (idx >= nE * 8) return;
    int e = idx >> 3, h = idx & 7;
    int t = it[e];
    float ex = __expf(scores[idx] - mx[t * 8 + h]);
    scores[idx] = ex;
    atomicAdd(sm + t * 8 + h, ex);
}

// Pass 3: one wave per edge; lane covers 4 of 128 dims of gated-V; scatter-add.
__global__ void edge_message(const float* __restrict__ expS, const float* __restrict__ sm,
                             const int* __restrict__ it, const int* __restrict__ is,
                             const float* __restrict__ KV, float* agg, int nE)
{
    int e = (int)((blockIdx.x * (size_t)blockDim.x + threadIdx.x) >> 5);
    int lane = threadIdx.x & 31;
    if (e >= nE) return;
    int t = it[e], s = is[e];
    int h = lane >> 2;                                   // 16 dims/head over 4 lanes
    float a = expS[(size_t)e * 8 + h] / (sm[t * 8 + h] + 1e-12f);
    float4 gv = *(const float4*)(KV + (size_t)s * 256 + 128 + lane * 4);  // gated V
    float* dst = agg + (size_t)t * 128 + lane * 4;
    atomicAdd(dst + 0, a * gv.x);
    atomicAdd(dst + 1, a * gv.y);
    atomicAdd(dst + 2, a * gv.z);
    atomicAdd(dst + 3, a * gv.w);
}

extern "C" void kernel_launch(void* const* d_in, const int* in_sizes, int n_in,
                              void* d_out, int out_size, void* d_ws, size_t ws_size,
                              hipStream_t stream)
{
    const float* X_tgt = (const float*)d_in[0];
    const float* X_src = (const float*)d_in[1];
    const int*   inc_t = (const int*)d_in[2];
    const int*   inc_s = (const int*)d_in[3];
    /* d_in[4] = num_tgt scalar: derived from in_sizes instead */
    const float* Wq    = (const float*)d_in[5];
    const float* Wkv   = (const float*)d_in[6];
    const float* Wbt   = (const float*)d_in[7];
    const float* Wbs   = (const float*)d_in[8];
    const float* Wgt_w = (const float*)d_in[9];
    const float* Wgt_b = (const float*)d_in[10];
    const float* Wgs_w = (const float*)d_in[11];
    const float* Wgs_b = (const float*)d_in[12];
    const float* Wo    = (const float*)d_in[13];

    const int DM = 128, H = 8;
    const int nT = in_sizes[0] / DM;
    const int nS = in_sizes[1] / DM;
    const int nE = in_sizes[2];

    // Workspace carve (~108 MB)
    float* ws     = (float*)d_ws;
    float* Q      = ws;  ws += (size_t)nT * DM;       // reused as gated activations
    float* KV     = ws;  ws += (size_t)nS * 2 * DM;   // [K | V] per row; V gated in place
    float* scores = ws;  ws += (size_t)nE * H;        // raw scores -> exp(scores)
    float* agg    = ws;  ws += (size_t)nT * DM;       // segment-summed messages
    float* bt     = ws;  ws += (size_t)nT * H;
    float* bs     = ws;  ws += (size_t)nS * H;
    float* mx     = ws;  ws += (size_t)nT * H;
    float* sm     = ws;  ws += (size_t)nT * H;

    dim3 blk(128);
    const int RT = 4;                                  // 64 rows per block

    // Init segment stats / accumulator
    {
        int n = nT * H;
        fill_f32<<<(n + 255) / 256, 256, 0, stream>>>(mx, -INFINITY, n);
        fill_f32<<<(n + 255) / 256, 256, 0, stream>>>(sm, 0.0f, n);
        int na = nT * DM;
        fill_f32<<<(na + 255) / 256, 256, 0, stream>>>(agg, 0.0f, na);
    }

    // Q = X_tgt @ Wq
    gemm_wmma<128,128,128,128,128,0,RT>
        <<<dim3((nT + 16*RT - 1) / (16*RT), (DM + 63) / 64), blk, 0, stream>>>(
            X_tgt, Wq, nullptr, Q, nullptr, nT, DM);

    // KV = X_src @ Wkv   (row: K[0:128] | V[128:256])
    gemm_wmma<128,128,256,256,256,0,RT>
        <<<dim3((nS + 16*RT - 1) / (16*RT), (2*DM + 63) / 64), blk, 0, stream>>>(
            X_src, Wkv, nullptr, KV, nullptr, nS, 2 * DM);

    // V *= sigmoid(X_src @ Wgs_w + Wgs_b)   (gating fused, in place on V half)
    gemm_wmma<128,128,128,256,256,1,RT>
        <<<dim3((nS + 16*RT - 1) / (16*RT), (DM + 63) / 64), blk, 0, stream>>>(
            X_src, Wgs_w, Wgs_b, KV + DM, KV + DM, nS, DM);

    // Head bias projections
    head_bias_gemm<<<(nT * H + 255) / 256, 256, 0, stream>>>(X_tgt, Wbt, bt, nT);
    head_bias_gemm<<<(nS * H + 255) / 256, 256, 0, stream>>>(X_src, Wbs, bs, nS);

    // Edge pass 1: scores + segment max
    {
        float scale = 1.0f / sqrtf((float)(DM / H));
        int n = nE * H;
        edge_scores<<<(n + 255) / 256, 256, 0, stream>>>(
            Q, KV, bt, bs, inc_t, inc_s, scores, mx, nE, scale);
    }
    // Edge pass 2: exp + segment sum
    {
        int n = nE * H;
        edge_exp<<<(n + 255) / 256, 256, 0, stream>>>(scores, mx, inc_t, sm, nE);
    }
    // Edge pass 3: message scatter-add (one wave32 per edge)
    edge_message<<<(nE + 7) / 8, 256, 0, stream>>>(
        scores, sm, inc_t, inc_s, KV, agg, nE);

    // gated = sigmoid(X_tgt @ Wgt_w + Wgt_b) * agg   (overwrites Q buffer)
    gemm_wmma<128,128,128,128,128,1,RT>
        <<<dim3((nT + 16*RT - 1) / (16*RT), (DM + 63) / 64), blk, 0, stream>>>(
            X_tgt, Wgt_w, Wgt_b, Q, agg, nT, DM);

    // out = gated @ Wo
    gemm_wmma<128,128,128,128,128,0,RT>
        <<<dim3((nT + 16*RT - 1) / (16*RT), (DM + 63) / 64), blk, 0, stream>>>(
            Q, Wo, nullptr, (float*)d_out, nullptr, nT, DM);
}